// Classifier_51934744543559
// MI455X (gfx1250) — compile-verified
//
#include <hip/hip_runtime.h>
#include <math.h>

typedef __attribute__((ext_vector_type(2))) float v2f;
typedef __attribute__((ext_vector_type(8))) float v8f;

#define NUM_CLASSES   10
#define DIM_CLASSES   64
#define DIM_TIMECAPS  16
#define NUM_TIMECAPS  512
#define KD            (NUM_CLASSES * DIM_CLASSES)   // 640
#define UPITCH        65          // 64 + 1 pad -> column walks hit all 64 LDS banks

// One workgroup per (batch, class) pair. 256 threads = 8 wave32.
// U tile for this (b,k) lives entirely in LDS: 512 x 64 f32 (pitch 65) = 130 KB
// out of the WGP's 320 KB -- never touches HBM.
__launch_bounds__(256, 1)
__global__ void capsule_route_kernel(const float* __restrict__ timecaps,  // [64,512,16]
                                     const float* __restrict__ conv_w,    // [16,1,640]
                                     const float* __restrict__ conv_b,    // [640]
                                     const float* __restrict__ B_bias,    // [10,1,512]
                                     float* __restrict__ out)             // [64,10,1]
{
    __shared__ float U[NUM_TIMECAPS * UPITCH];   // 133,120 B
    __shared__ float Usum[DIM_CLASSES];
    __shared__ float ae[NUM_TIMECAPS];           // agreements -> exp weights
    __shared__ float red[256];
    __shared__ float Svec[DIM_CLASSES];

    const int tid  = threadIdx.x;
    const int lane = tid & 31;
    const int wave = tid >> 5;
    const int half = lane >> 4;   // which 16-lane half of the wave
    const int l16  = lane & 15;

    const int bk = blockIdx.x;
    const int b  = bk / NUM_CLASSES;
    const int k  = bk % NUM_CLASSES;

    if (tid < DIM_CLASSES) Usum[tid] = 0.0f;
    __syncthreads();

    // ---------------- Pass 1: U = relu(timecaps[b] @ W_k + b_k) ----------------
    // V_WMMA_F32_16X16X4_F32, K=16 done as 4 chained k-steps.
    // B fragments (shared across all row-tiles): 4 k-steps x 4 col-tiles.
    v2f bfrag[4][4];
    const float* Wk = conv_w + k * DIM_CLASSES;   // row stride = 640 floats
    #pragma unroll
    for (int s = 0; s < 4; ++s) {
        const int kk = 4 * s + 2 * half;          // ISA B layout: VGPR0 holds K=kk, VGPR1 K=kk+1
        #pragma unroll
        for (int ct = 0; ct < 4; ++ct) {
            const int col = ct * 16 + l16;
            bfrag[s][ct].x = Wk[(kk + 0) * KD + col];
            bfrag[s][ct].y = Wk[(kk + 1) * KD + col];
        }
    }
    float biasv[4];
    #pragma unroll
    for (int ct = 0; ct < 4; ++ct) biasv[ct] = conv_b[k * DIM_CLASSES + ct * 16 + l16];

    const float* T = timecaps + (size_t)b * NUM_TIMECAPS * DIM_TIMECAPS;

    for (int rt = 0; rt < 4; ++rt) {              // 8 waves x 4 = 32 row-tiles of 16
        const int row0 = (wave * 4 + rt) * 16;
        v2f afrag[4];
        #pragma unroll
        for (int s = 0; s < 4; ++s) {             // A layout: lane row=l16, K pair 4s+2*half
            const float* ap = T + (row0 + l16) * DIM_TIMECAPS + 4 * s + 2 * half;
            afrag[s].x = ap[0];
            afrag[s].y = ap[1];
        }
        #pragma unroll
        for (int ct = 0; ct < 4; ++ct) {
            v8f acc = {};
            #pragma unroll
            for (int s = 0; s < 4; ++s) {
                acc = __builtin_amdgcn_wmma_f32_16x16x4_f32(
                        false, afrag[s], false, bfrag[s][ct],
                        (short)0, acc, false, false);
            }
            // D layout: VGPR r -> row row0+r+8*half, col ct*16+l16
            const int col = ct * 16 + l16;
            float csum = 0.0f;
            #pragma unroll
            for (int r = 0; r < 8; ++r) {
                float v = acc[r] + biasv[ct];
                v = v > 0.0f ? v : 0.0f;          // ReLU
                U[(row0 + r + 8 * half) * UPITCH + col] = v;
                csum += v;                        // partial column sum for Usum
            }
            atomicAdd(&Usum[col], csum);          // ds_add_f32
        }
    }
    __syncthreads();

    // ---------- Pass 2: a[m] = (Usum . U[m,:]) / sqrt(16)  (A.sum(-2) collapsed) ----------
    #pragma unroll
    for (int mm = 0; mm < 2; ++mm) {
        const int m = tid + mm * 256;
        float d = 0.0f;
        #pragma unroll 8
        for (int c = 0; c < DIM_CLASSES; ++c)
            d += Usum[c] * U[m * UPITCH + c];     // pitch-65: conflict-free
        ae[m] = d * 0.25f;
    }
    __syncthreads();

    // ---------- softmax over 512 agreements ----------
    red[tid] = fmaxf(ae[tid], ae[tid + 256]);
    __syncthreads();
    for (int off = 128; off > 0; off >>= 1) {
        if (tid < off) red[tid] = fmaxf(red[tid], red[tid + off]);
        __syncthreads();
    }
    const float amax = red[0];
    __syncthreads();

    const float e0 = __expf(ae[tid] - amax);
    const float e1 = __expf(ae[tid + 256] - amax);
    ae[tid] = e0;
    ae[tid + 256] = e1;
    red[tid] = e0 + e1;
    __syncthreads();
    for (int off = 128; off > 0; off >>= 1) {
        if (tid < off) red[tid] += red[tid + off];
        __syncthreads();
    }
    const float inv = 1.0f / red[0];

    // ---------- S[d] = sum_m (C[m] + B_bias[k,m]) * U[m,d] ----------
    const float* bb = B_bias + k * NUM_TIMECAPS;
    if (tid < DIM_CLASSES) {
        float s = 0.0f;
        for (int m = 0; m < NUM_TIMECAPS; ++m)
            s += (ae[m] * inv + bb[m]) * U[m * UPITCH + tid];
        Svec[tid] = s;
    }
    __syncthreads();

    // ---------- squash + output norm (closed form: v is a scalar multiple of S) ----------
    red[tid] = (tid < DIM_CLASSES) ? Svec[tid] * Svec[tid] : 0.0f;
    __syncthreads();
    for (int off = 128; off > 0; off >>= 1) {
        if (tid < off) red[tid] += red[tid + off];
        __syncthreads();
    }
    if (tid == 0) {
        const float n2  = red[0];
        const float nrm = sqrtf(n2);
        out[bk] = (n2 / (n2 + 1.0f)) * (nrm / (nrm + 1e-7f));
    }
}

extern "C" void kernel_launch(void* const* d_in, const int* in_sizes, int n_in,
                              void* d_out, int out_size, void* d_ws, size_t ws_size,
                              hipStream_t stream) {
    (void)in_sizes; (void)n_in; (void)out_size; (void)d_ws; (void)ws_size;
    const float* timecaps = (const float*)d_in[0];
    const float* conv_w   = (const float*)d_in[1];
    const float* conv_b   = (const float*)d_in[2];
    const float* B_bias   = (const float*)d_in[3];
    float* out = (float*)d_out;

    // 64 batches x 10 classes = 640 workgroups, 8 wave32 each
    capsule_route_kernel<<<dim3(64 * 10), dim3(256), 0, stream>>>(
        timecaps, conv_w, conv_b, B_bias, out);
}